// LSTMCell_33930241639040
// MI455X (gfx1250) — compile-verified
//
#include <hip/hip_runtime.h>

typedef float v2f __attribute__((ext_vector_type(2)));
typedef float v8f __attribute__((ext_vector_type(8)));

constexpr int BROWS  = 8192;   // batch
constexpr int DH     = 1024;   // hidden = input dim
constexpr int BM     = 128;    // block tile M
constexpr int BN     = 64;     // block tile N
constexpr int BK     = 32;     // K slice per stage
constexpr int LDT    = 36;     // padded LDS pitch (floats): conflict-free b64 frag reads
constexpr int ASZ    = BM * LDT;          // floats per A buffer
constexpr int BSZ    = BN * LDT;          // floats per B buffer
constexpr int NSLICE = 2 * (DH / BK);     // 2 GEMM phases (h|x) x 32 K-slices = 64

__device__ __forceinline__ float sigm_f(float z) {
  z = fminf(fmaxf(z, -30.f), 30.f);
  return 1.f / (1.f + __expf(-z));
}
__device__ __forceinline__ float tanh_f(float z) {
  z = fminf(fmaxf(z, -15.f), 15.f);
  const float e = __expf(-2.f * z);
  return (1.f - e) / (1.f + e);
}

// Wave-relative LDS byte offset of a generic pointer into __shared__ space.
__device__ __forceinline__ uint32_t lds_off(const void* p) {
  return (uint32_t)(uintptr_t)(__attribute__((address_space(3))) const void*)p;
}

// CDNA5 async copy: 16B global -> LDS, tracked by ASYNCcnt (no VGPR round-trip).
__device__ __forceinline__ void async_b128(uint32_t lds_byte, const float* g) {
  asm volatile("global_load_async_to_lds_b128 %0, %1, off"
               :
               : "v"(lds_byte), "v"(g)
               : "memory");
}

__launch_bounds__(256, 2)
__global__ void lstm_cell_wmma(const float* __restrict__ x,
                               const float* __restrict__ h,
                               const float* __restrict__ cin,
                               const float* __restrict__ Wh,
                               const float* __restrict__ Whb,
                               const float* __restrict__ Wx,
                               const float* __restrict__ Wxb,
                               float* __restrict__ out) {
  __shared__ float As[2][ASZ];   // double-buffered activation tiles 128x32 (padded)
  __shared__ float Bs[2][BSZ];   // double-buffered weight tiles      64x32 (padded)

  const int tid   = threadIdx.x;
  const int lane  = tid & 31;
  const int wave  = tid >> 5;
  const int waveM = (wave >> 1) << 5;   // 0,32,64,96
  const int waveN = (wave & 1) << 5;    // 0,32

  const int mBlock = blockIdx.y * BM;
  const int nBlock = blockIdx.x * BN;

  // global->LDS staging coordinates (each thread moves 6 x 16B per slice)
  const int lr = tid >> 3;            // 0..31 (row within slab)
  const int lc = (tid & 7) << 2;      // 0,4,...,28 (col, 16B aligned)

  // WMMA fragment coordinates
  const int nl = lane & 15;           // M / N index within 16
  const int kh = (lane >> 4) << 1;    // K sub-offset: 0 (lanes 0-15) or 2 (lanes 16-31)

  const uint32_t as0 = lds_off(&As[0][0]);
  const uint32_t bs0 = lds_off(&Bs[0][0]);

  // Issue the 6 async b128 copies for K-slice s into LDS buffer `buf`.
  auto issue = [&](int s, int buf) {
    const float* A  = (s < DH / BK) ? h : x;    // activations [8192,1024]
    const float* W  = (s < DH / BK) ? Wh : Wx;  // weights     [N,K] (=> @W^T free)
    const int    kb = (s & (DH / BK - 1)) << 5;
    const float* a  = A + (size_t)(mBlock + lr) * DH + kb + lc;
    const float* w  = W + (size_t)(nBlock + lr) * DH + kb + lc;
    const uint32_t ab = as0 + (uint32_t)(buf * ASZ + lr * LDT + lc) * 4u;
    const uint32_t bb = bs0 + (uint32_t)(buf * BSZ + lr * LDT + lc) * 4u;
    async_b128(ab,                    a);
    async_b128(ab + 32u * LDT * 4u,   a + (size_t)32 * DH);
    async_b128(ab + 64u * LDT * 4u,   a + (size_t)64 * DH);
    async_b128(ab + 96u * LDT * 4u,   a + (size_t)96 * DH);
    async_b128(bb,                    w);
    async_b128(bb + 32u * LDT * 4u,   w + (size_t)32 * DH);
  };

  v8f acc[2][2] = {};

  issue(0, 0);                                  // prologue: slice 0 -> buf 0

  #pragma unroll 1
  for (int i = 0; i < NSLICE; ++i) {
    const int cur = i & 1;

    if (i + 1 < NSLICE) {
      issue(i + 1, cur ^ 1);                    // producer for next slice
      // async loads retire in order: <=6 outstanding => slice i fully landed
      asm volatile("s_wait_asynccnt 0x6" ::: "memory");
    } else {
      asm volatile("s_wait_asynccnt 0x0" ::: "memory");
    }
    __syncthreads();                            // slice i visible to all waves

    const float* as = &As[cur][0];
    const float* bs = &Bs[cur][0];
    #pragma unroll
    for (int kk = 0; kk < BK; kk += 4) {
      const int ko = kk + kh;
      const v2f a0 = *(const v2f*)&as[(waveM +      nl) * LDT + ko];
      const v2f a1 = *(const v2f*)&as[(waveM + 16 + nl) * LDT + ko];
      const v2f b0 = *(const v2f*)&bs[(waveN +      nl) * LDT + ko];
      const v2f b1 = *(const v2f*)&bs[(waveN + 16 + nl) * LDT + ko];
      acc[0][0] = __builtin_amdgcn_wmma_f32_16x16x4_f32(false, a0, false, b0,
                                                        (short)0, acc[0][0], false, false);
      acc[0][1] = __builtin_amdgcn_wmma_f32_16x16x4_f32(false, a0, false, b1,
                                                        (short)0, acc[0][1], false, false);
      acc[1][0] = __builtin_amdgcn_wmma_f32_16x16x4_f32(false, a1, false, b0,
                                                        (short)0, acc[1][0], false, false);
      acc[1][1] = __builtin_amdgcn_wmma_f32_16x16x4_f32(false, a1, false, b1,
                                                        (short)0, acc[1][1], false, false);
    }
    __syncthreads();                            // buf cur free before iter i+1 refills it
  }

  // ---- fused LSTM epilogue: z -> (h_new, c_new), no z round-trip ----
  const int   n0    = nBlock + waveN + nl;
  const int   n1    = n0 + 16;
  const float bias0 = Whb[n0] + Wxb[n0];
  const float bias1 = Whb[n1] + Wxb[n1];
  const int   mh    = (lane >> 4) << 3;    // C/D layout: lanes 16-31 hold rows M+8

  float* __restrict__ hout = out;
  float* __restrict__ cout = out + (size_t)BROWS * DH;

  #pragma unroll
  for (int mi = 0; mi < 2; ++mi) {
    #pragma unroll
    for (int i = 0; i < 8; ++i) {
      const int    m   = mBlock + waveM + mi * 16 + mh + i;
      const size_t row = (size_t)m * DH;
      #pragma unroll
      for (int ni = 0; ni < 2; ++ni) {
        const int   n  = ni ? n1 : n0;
        const float z  = acc[mi][ni][i] + (ni ? bias1 : bias0);
        const float s  = sigm_f(z);          // i = f = o gates
        const float g  = tanh_f(z);          // candidate
        const float c0 = cin[row + n];
        const float cn = s * (c0 + g);       // s*c + s*g
        const float hn = s * tanh_f(cn);
        hout[row + n] = hn;
        cout[row + n] = cn;
      }
    }
  }
}

extern "C" void kernel_launch(void* const* d_in, const int* in_sizes, int n_in,
                              void* d_out, int out_size, void* d_ws, size_t ws_size,
                              hipStream_t stream) {
  const float* x    = (const float*)d_in[0];
  const float* h    = (const float*)d_in[1];
  const float* c    = (const float*)d_in[2];
  const float* Wh_w = (const float*)d_in[3];
  const float* Wh_b = (const float*)d_in[4];
  const float* Wx_w = (const float*)d_in[5];
  const float* Wx_b = (const float*)d_in[6];
  (void)in_sizes; (void)n_in; (void)out_size; (void)d_ws; (void)ws_size;

  dim3 grid(DH / BN, BROWS / BM);   // 16 x 64 blocks
  lstm_cell_wmma<<<grid, 256, 0, stream>>>(x, h, c, Wh_w, Wh_b, Wx_w, Wx_b,
                                           (float*)d_out);
}